// CF2DConv_11553462026517
// MI455X (gfx1250) — compile-verified
//
#include <hip/hip_runtime.h>
#include <hip/hip_bf16.h>

// CF2DConv (FNO spectral conv) for gfx1250, fp32 WMMA (V_WMMA_F32_16X16X4_F32)
// with async global->LDS tile staging (GLOBAL_LOAD_ASYNC_TO_LDS_*, ASYNCcnt)
// and double-buffered K-tiles.
//
// Pipeline (modes truncated to 48x48, so all transforms shrink to 48-row GEMMs):
//  S1: T1[b,x,ky,c]  = sum_y  Dy[ky,y]  * x[b,x,y,c]           (DCT-II ortho, keep 48)
//  S2: T2r/i[b,kx,ky,c] = sum_x Fr/Fi[kx,x] * T1[b,x,ky,c]     (rFFT ortho, keep 48)
//  S3: per mode (a,b): Y[i,e] = sum_j R[i,j]*X[e,j]  (complex) (channel mixing)
//  S4: T4[b,x,ky,c]  = sum_kx Gr[x,kx]*T3r + Gi[x,kx]*T3i      (irfft ortho from 48)
//  S5: out[b,x,y,c]  = sum_ky Dy[ky,y]  * T4[b,x,ky,c]         (iDCT ortho from 48)

typedef float v2f __attribute__((ext_vector_type(2)));
typedef float v4f __attribute__((ext_vector_type(4)));
typedef float v8f __attribute__((ext_vector_type(8)));

__device__ __forceinline__ v8f wmma4(v2f a, v2f b, v8f c) {
  // D(16x16 f32) = A(16x4 f32) * B(4x16 f32) + C
  return __builtin_amdgcn_wmma_f32_16x16x4_f32(false, a, false, b, (short)0, c,
                                               false, false);
}

// Low 32 bits of a generic pointer into LDS == LDS byte offset (flat aperture
// maps LDS as addr[31:0]).
__device__ __forceinline__ unsigned lds_off(const void* p) {
  return (unsigned)(uintptr_t)p;
}

__device__ __forceinline__ void async_b128(unsigned ldsdst, const void* gsrc) {
  asm volatile("global_load_async_to_lds_b128 %0, %1, off"
               :: "v"(ldsdst), "v"(gsrc) : "memory");
}
__device__ __forceinline__ void async_b32(unsigned ldsdst, const void* gsrc) {
  asm volatile("global_load_async_to_lds_b32 %0, %1, off"
               :: "v"(ldsdst), "v"(gsrc) : "memory");
}
__device__ __forceinline__ void wait_async0() {
  asm volatile("s_wait_asynccnt 0x0" ::: "memory");
}

// ---------------------------------------------------------------------------
// Transform-matrix generation (integer angle reduction -> exact periodicity)
// ---------------------------------------------------------------------------
__global__ void init_mats(float* __restrict__ Dy, float* __restrict__ Fr,
                          float* __restrict__ Fi, float* __restrict__ DyT,
                          float* __restrict__ G) {
  const float PI = 3.14159265358979323846f;
  int idx = blockIdx.x * blockDim.x + threadIdx.x;
  if (idx < 48 * 256) {
    int k = idx >> 8, n = idx & 255;
    // Orthonormal DCT-II: Dy[k,n] = s_k * cos(pi*(2n+1)*k / 512)
    float s = (k == 0) ? 0.0625f : 0.0883883476483184f;  // sqrt(1/256), sqrt(2/256)
    int mm = (k * (2 * n + 1)) & 1023;                   // period 1024
    float d = s * cosf(PI * (float)mm * (1.0f / 512.0f));
    Dy[idx] = d;
    DyT[n * 48 + k] = d;
    // Ortho rFFT (keep 48): F[k,n] = exp(-2*pi*i*k*n/256)/16
    int t = (k * n) & 255;
    float ang = (2.0f * PI) * (float)t * (1.0f / 256.0f);
    Fr[idx] = cosf(ang) * 0.0625f;
    Fi[idx] = -sinf(ang) * 0.0625f;
  }
  if (idx < 256 * 96) {
    // Ortho irfft from 48 modes (k=0 unique, k>=1 doubled, no Nyquist):
    // Gr[n,k] = (k==0?1:2cos)/16 ; Gi[n,k] = (k==0?0:-2sin)/16
    int n = idx / 96, kk = idx % 96;
    const float PI2 = 2.0f * PI;
    float g;
    if (kk < 48) {
      int k = kk;
      g = (k == 0) ? 0.0625f
                   : 2.0f * cosf(PI2 * (float)((k * n) & 255) * (1.0f / 256.0f)) * 0.0625f;
    } else {
      int k = kk - 48;
      g = (k == 0) ? 0.0f
                   : -2.0f * sinf(PI2 * (float)((k * n) & 255) * (1.0f / 256.0f)) * 0.0625f;
    }
    G[idx] = g;
  }
}

// ---------------------------------------------------------------------------
// Generic fp32 WMMA GEMM: C(M x 128) = A(M x K) * B(K x 128), slab-batched.
// Block = 128 threads (4 waves); block tile = 16(M) x 128(N); K-tile = 16.
// Async global->LDS staging, double-buffered. B rows may come from two
// buffers split at Ksplit (stage-4 concat).
// ---------------------------------------------------------------------------
#define LDA_P 20   // 16-col A tile padded: bank = (20m+k)%64 distinct over m
#define LDB_P 132  // 128-col B tile padded, 16B-aligned rows
#define ATILE (16 * LDA_P)
#define BTILE (16 * LDB_P)

__global__ __launch_bounds__(128) void gemm_wmma(
    const float* __restrict__ A, int ldA, int K, int Ksplit,
    const float* __restrict__ B1, const float* __restrict__ B2,
    long long strB1, long long strB2, int ldB, int S2,
    float* __restrict__ C, long long strC1, long long strC2, int ldC) {
  __shared__ __align__(16) float Als[2 * ATILE];
  __shared__ __align__(16) float Bls[2 * BTILE];

  const int slab = blockIdx.x;
  const int s1 = slab / S2, s2 = slab % S2;
  const long long bOff = (long long)s1 * strB1 + (long long)s2 * strB2;
  const float* Bb1 = B1 + bOff;
  const float* Bb2 = B2 + bOff;
  float* Cb = C + (long long)s1 * strC1 + (long long)s2 * strC2;
  const int m0 = blockIdx.y * 16;

  const int tid = threadIdx.x;
  const int wave = tid >> 5, lane = tid & 31;
  const int half = lane >> 4, m = lane & 15;
  const int n0 = wave * 32;  // each wave: two 16-wide N tiles

  auto issue_tile = [&](int buf, int kt) {
    if (tid < 64) {  // A tile 16x16, one b128 per thread
      int r = tid >> 2, c = (tid & 3) << 2;
      async_b128(lds_off(&Als[buf * ATILE + r * LDA_P + c]),
                 &A[(long long)(m0 + r) * ldA + (kt + c)]);
    }
#pragma unroll
    for (int p = 0; p < 4; ++p) {  // B tile 16x128, four b128 per thread
      int r = (tid >> 5) + p * 4;
      int c = (tid & 31) << 2;
      int gk = kt + r;
      const float* src = (gk < Ksplit) ? (Bb1 + (long long)gk * ldB + c)
                                       : (Bb2 + (long long)(gk - Ksplit) * ldB + c);
      async_b128(lds_off(&Bls[buf * BTILE + r * LDB_P + c]), src);
    }
  };

  v8f acc0 = {}, acc1 = {};
  issue_tile(0, 0);
  int cur = 0;

  for (int kt = 0; kt < K; kt += 16) {
    wait_async0();      // my async loads into buf[cur] have landed in LDS
    __syncthreads();    // everyone's landed; buf[cur^1] fully consumed last iter
    if (kt + 16 < K) issue_tile(cur ^ 1, kt + 16);  // overlap with compute

    const float* Al = &Als[cur * ATILE];
    const float* Bl = &Bls[cur * BTILE];
#pragma unroll
    for (int ks = 0; ks < 4; ++ks) {
      // ISA 16x4 A layout: lanes<16 hold K=kk,kk+1 ; lanes>=16 hold kk+2,kk+3
      int kk = ks * 4 + half * 2;
      v2f a = *(const v2f*)&Al[m * LDA_P + kk];
      v2f b0, b1;
      b0.x = Bl[kk * LDB_P + n0 + m];
      b0.y = Bl[(kk + 1) * LDB_P + n0 + m];
      b1.x = Bl[kk * LDB_P + n0 + 16 + m];
      b1.y = Bl[(kk + 1) * LDB_P + n0 + 16 + m];
      acc0 = wmma4(a, b0, acc0);
      acc1 = wmma4(a, b1, acc1);
    }
    cur ^= 1;
  }

  // C layout: VGPR r -> row r (lanes<16) / row r+8 (lanes>=16), col = lane&15
#pragma unroll
  for (int r = 0; r < 8; ++r) {
    int row = m0 + r + half * 8;
    Cb[(long long)row * ldC + (n0 + m)] = acc0[r];
    Cb[(long long)row * ldC + (n0 + 16 + m)] = acc1[r];
  }
}

// ---------------------------------------------------------------------------
// Stage 3: per-mode complex channel mixing.
// C[i,e] = sum_j R[i,j] * X[e,j] ; Yr = Rr*Xr - Ri*Xi ; Yi = Rr*Xi + Ri*Xr
// M = 128 (i, 4 waves x 2 tiles), N = 16 (e padded from 8), K = 128 (j).
// X^T staged in LDS (async b32 scatters) as (j-pair, e)-interleaved pairs ->
// conflict-free ds_load_b64 B-fragments. R fragments read directly from
// global (each element used exactly once; dominant streaming traffic).
// ---------------------------------------------------------------------------
__global__ __launch_bounds__(128) void mode_kernel(
    const float* __restrict__ Rr, const float* __restrict__ Ri,
    const float* __restrict__ Xr, const float* __restrict__ Xi,
    float* __restrict__ Yr, float* __restrict__ Yi) {
  __shared__ __align__(16) float XrT[64 * 32];  // idx = (j>>1)*32 + e*2 + (j&1)
  __shared__ __align__(16) float XiT[64 * 32];

  const int mode = blockIdx.x;                     // a*48 + b
  const long long roff = (long long)mode * 16384;  // 128*128 per mode
  const int xoff = mode * 128;                     // + e*294912 + j
  const int tid = threadIdx.x;                     // == j for staging
  const int wave = tid >> 5, lane = tid & 31;
  const int half = lane >> 4, m = lane & 15;

  const int sidx = (tid >> 1) * 32 + (tid & 1);
#pragma unroll
  for (int e = 0; e < 8; ++e) {  // async per-lane scatters, coalesced over j
    async_b32(lds_off(&XrT[sidx + e * 2]), &Xr[(long long)e * 294912 + xoff + tid]);
    async_b32(lds_off(&XiT[sidx + e * 2]), &Xi[(long long)e * 294912 + xoff + tid]);
  }
#pragma unroll
  for (int e = 8; e < 16; ++e) {  // zero-pad batch 8 -> 16 (plain DS stores)
    XrT[sidx + e * 2] = 0.0f;
    XiT[sidx + e * 2] = 0.0f;
  }
  wait_async0();
  __syncthreads();

  const int mt0 = wave * 32;  // this wave's i range: two 16-row tiles
  v8f aR0 = {}, aS0 = {}, aI0 = {}, aR1 = {}, aS1 = {}, aI1 = {};

  for (int ks = 0; ks < 32; ++ks) {
    int kk = ks * 4 + half * 2;
    v2f bxr = *(const v2f*)&XrT[(kk >> 1) * 32 + m * 2];
    v2f bxi = *(const v2f*)&XiT[(kk >> 1) * 32 + m * 2];
    v2f ar0 = *(const v2f*)&Rr[roff + (long long)(mt0 + m) * 128 + kk];
    v2f ai0 = *(const v2f*)&Ri[roff + (long long)(mt0 + m) * 128 + kk];
    v2f ar1 = *(const v2f*)&Rr[roff + (long long)(mt0 + 16 + m) * 128 + kk];
    v2f ai1 = *(const v2f*)&Ri[roff + (long long)(mt0 + 16 + m) * 128 + kk];
    aR0 = wmma4(ar0, bxr, aR0);  // Rr*Xr
    aS0 = wmma4(ai0, bxi, aS0);  // Ri*Xi (subtract at end; f32 WMMA has no A-neg)
    aI0 = wmma4(ar0, bxi, aI0);  // Rr*Xi
    aI0 = wmma4(ai0, bxr, aI0);  // + Ri*Xr
    aR1 = wmma4(ar1, bxr, aR1);
    aS1 = wmma4(ai1, bxi, aS1);
    aI1 = wmma4(ar1, bxi, aI1);
    aI1 = wmma4(ai1, bxr, aI1);
  }

  if (m < 8) {  // only real batches; divergence is after all WMMAs
    long long ebase = (long long)m * 294912 + xoff;
#pragma unroll
    for (int r = 0; r < 8; ++r) {
      int i0 = mt0 + r + half * 8;
      int i1 = mt0 + 16 + r + half * 8;
      Yr[ebase + i0] = aR0[r] - aS0[r];
      Yi[ebase + i0] = aI0[r];
      Yr[ebase + i1] = aR1[r] - aS1[r];
      Yi[ebase + i1] = aI1[r];
    }
  }
}

// ---------------------------------------------------------------------------
extern "C" void kernel_launch(void* const* d_in, const int* in_sizes, int n_in,
                              void* d_out, int out_size, void* d_ws,
                              size_t ws_size, hipStream_t stream) {
  const float* x  = (const float*)d_in[0];  // (8,256,256,128) f32
  const float* Rr = (const float*)d_in[1];  // (48,48,128,128) f32: per-mode 128x128
  const float* Ri = (const float*)d_in[2];
  float* out = (float*)d_out;               // (8,256,256,128) f32
  float* ws = (float*)d_ws;

  // Workspace layout (floats); total ~138.7 MB
  float* Dy  = ws;              // 48*256
  float* Fr  = ws + 12288;      // 48*256
  float* Fi  = ws + 24576;      // 48*256
  float* DyT = ws + 36864;      // 256*48
  float* G   = ws + 49152;      // 256*96  = [Gr | Gi]
  float* T1  = ws + 73728;      // (8,256,48,128)
  float* T2r = T1 + 12582912;   // (8,48,48,128)
  float* T2i = T2r + 2359296;
  float* T3r = T2i + 2359296;
  float* T3i = T3r + 2359296;
  float* T4  = T3i + 2359296;   // (8,256,48,128)

  init_mats<<<96, 256, 0, stream>>>(Dy, Fr, Fi, DyT, G);

  // S1: per (b,x) slab: T1 = Dy(48x256) * x(256x128)
  gemm_wmma<<<dim3(2048, 3), 128, 0, stream>>>(
      Dy, 256, 256, 256, x, x, 32768LL, 0LL, 128, 1, T1, 6144LL, 0LL, 128);

  // S2: per (b,ky): T2r = Fr(48x256) * T1(256x128) ; T2i = Fi * T1
  gemm_wmma<<<dim3(8 * 48, 3), 128, 0, stream>>>(
      Fr, 256, 256, 256, T1, T1, 1572864LL, 128LL, 6144, 48,
      T2r, 294912LL, 128LL, 6144);
  gemm_wmma<<<dim3(8 * 48, 3), 128, 0, stream>>>(
      Fi, 256, 256, 256, T1, T1, 1572864LL, 128LL, 6144, 48,
      T2i, 294912LL, 128LL, 6144);

  // S3: 2304 modes of complex 128x128 channel mixing
  mode_kernel<<<2304, 128, 0, stream>>>(Rr, Ri, T2r, T2i, T3r, T3i);

  // S4: per (b,ky): T4 = [Gr|Gi](256x96) * [T3r;T3i](96x128)  (K split at 48)
  gemm_wmma<<<dim3(8 * 48, 16), 128, 0, stream>>>(
      G, 96, 96, 48, T3r, T3i, 294912LL, 128LL, 6144, 48,
      T4, 1572864LL, 128LL, 6144);

  // S5: per (b,x): out = DyT(256x48) * T4(48x128)
  gemm_wmma<<<dim3(2048, 16), 128, 0, stream>>>(
      DyT, 48, 48, 48, T4, T4, 6144LL, 0LL, 128, 1, out, 32768LL, 0LL, 128);
}